// FlowmatchingActionHead_46205258170745
// MI455X (gfx1250) — compile-verified
//
#include <hip/hip_runtime.h>

typedef __attribute__((ext_vector_type(2))) float v2f;
typedef __attribute__((ext_vector_type(8))) float v8f;

#define NUM_B 128
#define SEQ_T 64
#define DIM_IN 1024
#define DIM_HID 1024
#define DIM_OUT 1536
#define N_CAT 32
#define NS 4                       // samples (of same category) per chunk
#define NCHUNK (NUM_B / NS)        // worst case: all samples in one category
#define WAVES_PER_BLOCK 4
#define COLS_PER_BLOCK (WAVES_PER_BLOCK * 16)

// ---------------------------------------------------------------------------
// Build per-category sample lists (deterministic, serial; trivial work).
// counts[c] = #samples with cat c; lists[c*NUM_B + i] = i-th such sample id.
// ---------------------------------------------------------------------------
__global__ void build_lists_kernel(const int* __restrict__ cat_ids,
                                   int* __restrict__ counts,
                                   int* __restrict__ lists) {
  if (threadIdx.x == 0 && blockIdx.x == 0) {
    for (int c = 0; c < N_CAT; ++c) counts[c] = 0;
    for (int b = 0; b < NUM_B; ++b) {
      int c = cat_ids[b];
      lists[c * NUM_B + counts[c]] = b;
      counts[c] = counts[c] + 1;
    }
  }
}

// ---------------------------------------------------------------------------
// Grouped expert GEMM:  Y[bs] = act( X[bs] @ W[cat] + bias[cat] )
// One block: NS same-category samples (M = NS*64 rows) x 64 columns, K-loop.
// Wave tile: 16 cols x 16 M-subtiles via V_WMMA_F32_16X16X4_F32.
// ---------------------------------------------------------------------------
template <int K, int N, bool RELU>
__global__ __launch_bounds__(WAVES_PER_BLOCK * 32)
void expert_gemm_kernel(const float* __restrict__ X,     // (NUM_B, SEQ_T, K)
                        const float* __restrict__ W,     // (N_CAT, K, N)
                        const float* __restrict__ bias,  // (N_CAT, N)
                        float* __restrict__ Y,           // (NUM_B, SEQ_T, N)
                        const int* __restrict__ lists,
                        const int* __restrict__ counts) {
  const int cat = blockIdx.y;
  const int chunk = blockIdx.z;
  const int cnt = counts[cat];
  const int s0 = chunk * NS;
  if (s0 >= cnt) return;  // over-launched chunk: nothing to do

  const int wave = threadIdx.x >> 5;
  const int lane = threadIdx.x & 31;
  const int l15 = lane & 15;
  const int hi = lane >> 4;  // 0: lanes 0-15, 1: lanes 16-31

  const int col = blockIdx.x * COLS_PER_BLOCK + wave * 16 + l15;

  int bs[NS];
  bool valid[NS];
#pragma unroll
  for (int s = 0; s < NS; ++s) {
    int idx = s0 + s;
    valid[s] = idx < cnt;
    bs[s] = lists[cat * NUM_B + (valid[s] ? idx : s0)];  // dup sample 0 if tail
  }

  const float* Wc = W + (size_t)cat * K * N;
  // B fragment: v0 = W[k + 2*hi][col], v1 = W[k + 2*hi + 1][col]
  const float* wp = Wc + (size_t)(2 * hi) * N + col;

  // A row pointers: 16 M-subtiles = NS samples x 4 sub-tiles of 16 rows.
  // A fragment per lane: float2 {X[row][k + 2*hi], X[row][k + 2*hi + 1]}
  const float* ap[4 * NS];
#pragma unroll
  for (int m = 0; m < 4 * NS; ++m) {
    int s = m >> 2;
    int mt = m & 3;
    ap[m] = X + ((size_t)bs[s] * SEQ_T + mt * 16 + l15) * K + 2 * hi;
  }

  v8f acc[4 * NS] = {};

  for (int k = 0; k < K; k += 4) {
    v2f bfrag;
    bfrag[0] = wp[(size_t)k * N];
    bfrag[1] = wp[(size_t)k * N + N];
#pragma unroll
    for (int m = 0; m < 4 * NS; ++m) {
      v2f afrag = *(const v2f*)(ap[m] + k);
      acc[m] = __builtin_amdgcn_wmma_f32_16x16x4_f32(
          /*neg_a=*/false, afrag, /*neg_b=*/false, bfrag,
          /*c_mod=*/(short)0, acc[m], /*reuse_a=*/false, /*reuse_b=*/false);
    }
  }

  // Epilogue: bias (+ReLU) and store. D layout: VGPR r holds M = r (+8 if hi).
  const float bv = bias[cat * N + col];
#pragma unroll
  for (int m = 0; m < 4 * NS; ++m) {
    int s = m >> 2;
    if (!valid[s]) continue;
    int mt = m & 3;
    float* yb = Y + ((size_t)bs[s] * SEQ_T + mt * 16 + hi * 8) * N + col;
#pragma unroll
    for (int r = 0; r < 8; ++r) {
      float v = acc[m][r] + bv;
      if (RELU) v = fmaxf(v, 0.0f);
      yb[(size_t)r * N] = v;
    }
  }
}

// ---------------------------------------------------------------------------
extern "C" void kernel_launch(void* const* d_in, const int* in_sizes, int n_in,
                              void* d_out, int out_size, void* d_ws,
                              size_t ws_size, hipStream_t stream) {
  (void)in_sizes; (void)n_in; (void)out_size; (void)ws_size;

  const float* x = (const float*)d_in[0];        // (128, 64, 1024)
  const float* W1 = (const float*)d_in[1];       // (32, 1024, 1024)
  const float* b1 = (const float*)d_in[2];       // (32, 1024)
  const float* W2 = (const float*)d_in[3];       // (32, 1024, 1536)
  const float* b2 = (const float*)d_in[4];       // (32, 1536)
  const int* cat_ids = (const int*)d_in[5];      // (128,)
  float* out = (float*)d_out;                    // (128, 64, 1536)

  char* ws = (char*)d_ws;
  int* counts = (int*)ws;                        // 32 ints
  int* lists = (int*)(ws + 256);                 // 32*128 ints (16 KB)
  float* h = (float*)(ws + 65536);               // (128, 64, 1024) f32 = 32 MB

  build_lists_kernel<<<1, 32, 0, stream>>>(cat_ids, counts, lists);

  dim3 blk(WAVES_PER_BLOCK * 32);
  dim3 g1(DIM_HID / COLS_PER_BLOCK, N_CAT, NCHUNK);   // (16, 32, 32)
  expert_gemm_kernel<DIM_IN, DIM_HID, true>
      <<<g1, blk, 0, stream>>>(x, W1, b1, h, lists, counts);

  dim3 g2(DIM_OUT / COLS_PER_BLOCK, N_CAT, NCHUNK);   // (24, 32, 32)
  expert_gemm_kernel<DIM_HID, DIM_OUT, false>
      <<<g2, blk, 0, stream>>>(h, W2, b2, out, lists, counts);
}